// DNDEstimator_77489799955147
// MI455X (gfx1250) — compile-verified
//
#include <hip/hip_runtime.h>
#include <stdint.h>

typedef __attribute__((ext_vector_type(2))) float v2f;
typedef __attribute__((ext_vector_type(8))) float v8f;

#define B_ROWS 1024
#define S_DIM  256
#define H_DIM  64
#define A_DIM  18
#define N_MEM  200000
#define K_NB   50
#define DND_DELTA 1e-3f
#define NSPLIT 8
#define NPW    25000          // columns per split (200000 / 8)
#define NTILES 1563           // ceil(25000 / 16)
#define CAP    384            // pool capacity: 128 (pre-flush max) + 256 (per-iter max)
#define F_INF  __builtin_inff()

// ---------------------------------------------------------------------------
// Kernel 1: h = relu(x @ W1 + b1)   [1024,256]@[256,64] via f32 WMMA
// grid: 32 blocks * 256 threads; one wave per 16x16 output tile (64*4 tiles)
// ---------------------------------------------------------------------------
__global__ void __launch_bounds__(256) k_hidden(const float* __restrict__ x,
                                                const float* __restrict__ W1,
                                                const float* __restrict__ b1,
                                                float* __restrict__ h) {
  int tile = blockIdx.x * 8 + (threadIdx.x >> 5);
  int lane = threadIdx.x & 31;
  int mt = tile >> 2, nt = tile & 3;
  int r0 = mt * 16, n0 = nt * 16;
  int lr = lane & 15, hi = lane >> 4;

  v8f c = {};
  const float* aBase = x + (size_t)(r0 + lr) * S_DIM + 2 * hi;
  const float* bCol  = W1 + n0 + lr;
  for (int t = 0; t < S_DIM / 4; ++t) {
    v2f a = *(const v2f*)(aBase + 4 * t);          // x[r][4t+2hi .. +1]
    int k = 4 * t + 2 * hi;
    v2f b;
    b.x = bCol[(size_t)(k + 0) * H_DIM];           // W1[k][n]
    b.y = bCol[(size_t)(k + 1) * H_DIM];
    c = __builtin_amdgcn_wmma_f32_16x16x4_f32(false, a, false, b,
                                              (short)0, c, false, false);
  }
  float bias = b1[n0 + lr];
#pragma unroll
  for (int v = 0; v < 8; ++v) {
    int m = hi * 8 + v;
    float val = c[v] + bias;
    val = val > 0.f ? val : 0.f;
    h[(size_t)(r0 + m) * H_DIM + n0 + lr] = val;
  }
}

// ---------------------------------------------------------------------------
// Kernel 2: policy = h @ Wp + bp   (tiny: 1024x18, K=64)
// ---------------------------------------------------------------------------
__global__ void __launch_bounds__(256) k_policy(const float* __restrict__ h,
                                                const float* __restrict__ Wp,
                                                const float* __restrict__ bp,
                                                float* __restrict__ pol) {
  int i = blockIdx.x * blockDim.x + threadIdx.x;
  if (i >= B_ROWS * A_DIM) return;
  int r = i / A_DIM, a = i % A_DIM;
  float s = bp[a];
  const float* hr = h + (size_t)r * H_DIM;
#pragma unroll 8
  for (int k = 0; k < H_DIM; ++k) s += hr[k] * Wp[k * A_DIM + a];
  pol[i] = s;
}

// ---------------------------------------------------------------------------
// Kernel 3: fused distance GEMM + streaming per-row top-50.
// grid = 64 row-blocks * 8 N-splits = 512 blocks of 256 threads (8 waves).
// Each wave computes TWO independent 16x16 tiles of  s = h @ keys^T  per
// iteration (two interleaved v_wmma_f32_16x16x4_f32 accumulator chains to
// hide matrix-pipe latency), derives k_sq from the same B-fragment loads,
// and appends candidates p = k_sq - 2s (per-row-constant h_sq deferred)
// into a threshold-filtered LDS pool with exact top-50 flushes.
// ---------------------------------------------------------------------------
__global__ void __launch_bounds__(256) k_dnd_partial(
    const float* __restrict__ h, const float* __restrict__ keys,
    float* __restrict__ cand_d, unsigned* __restrict__ cand_i) {
  __shared__ float    s_pd[16 * CAP];
  __shared__ unsigned s_pi[16 * CAP];
  __shared__ unsigned s_cnt[16];
  __shared__ float    s_thr[16];

  int lane = threadIdx.x & 31;
  int wv   = threadIdx.x >> 5;
  int rowBlk = blockIdx.x >> 3;     // 0..63
  int split  = blockIdx.x & 7;      // 0..7
  int r0 = rowBlk * 16;
  int nStart = split * NPW;
  int lr = lane & 15, hi = lane >> 4;

  if (threadIdx.x < 16) { s_cnt[threadIdx.x] = 0u; s_thr[threadIdx.x] = F_INF; }

  // A fragments: h rows r0..r0+15, K=64 -> 16 fragments of 2 floats/lane
  v2f afrag[16];
  {
    const float* aBase = h + (size_t)(r0 + lr) * H_DIM + 2 * hi;
#pragma unroll
    for (int t = 0; t < 16; ++t) afrag[t] = *(const v2f*)(aBase + 4 * t);
  }
  __syncthreads();

  const int NPAIRS = (NTILES + 1) / 2;     // 782 tile-pairs
  const int iters  = (NPAIRS + 7) / 8;     // 98 barrier rounds
  for (int it = 0; it < iters; ++it) {
    int pair = it * 8 + wv;
    int tile0 = pair * 2;
    if (tile0 < NTILES) {
      int tile1 = tile0 + 1;
      int c0 = nStart + tile0 * 16 + lr;   // this lane's key row, chain 0
      int c1 = nStart + tile1 * 16 + lr;   // this lane's key row, chain 1
      bool valid0 = (c0 < nStart + NPW);
      bool valid1 = (tile1 < NTILES) && (c1 < nStart + NPW);
      int cld0 = valid0 ? c0 : (N_MEM - 1);
      int cld1 = valid1 ? c1 : (N_MEM - 1);
      const float* b0Base = keys + (size_t)cld0 * H_DIM + 2 * hi;
      const float* b1Base = keys + (size_t)cld1 * H_DIM + 2 * hi;

      v8f acc0 = {}, acc1 = {};
      float ksq0 = 0.f, ksq1 = 0.f;
#pragma unroll
      for (int t = 0; t < 16; ++t) {
        v2f b0 = *(const v2f*)(b0Base + 4 * t);
        v2f b1 = *(const v2f*)(b1Base + 4 * t);
        ksq0 += b0.x * b0.x + b0.y * b0.y;
        ksq1 += b1.x * b1.x + b1.y * b1.y;
        acc0 = __builtin_amdgcn_wmma_f32_16x16x4_f32(false, afrag[t], false, b0,
                                                     (short)0, acc0, false, false);
        acc1 = __builtin_amdgcn_wmma_f32_16x16x4_f32(false, afrag[t], false, b1,
                                                     (short)0, acc1, false, false);
      }
      ksq0 += __shfl_xor(ksq0, 16, 32);    // full row sum-of-squares
      ksq1 += __shfl_xor(ksq1, 16, 32);

      float thr[8];
#pragma unroll
      for (int v = 0; v < 8; ++v) thr[v] = s_thr[hi * 8 + v];

#pragma unroll
      for (int v = 0; v < 8; ++v) {
        int m = hi * 8 + v;                // row within 16-row block
        float p0 = valid0 ? (ksq0 - 2.f * acc0[v]) : F_INF;
        if (p0 < thr[v]) {
          unsigned pos = atomicAdd(&s_cnt[m], 1u);   // pos < CAP by construction
          s_pd[m * CAP + pos] = p0;
          s_pi[m * CAP + pos] = (unsigned)c0;
        }
        float p1 = valid1 ? (ksq1 - 2.f * acc1[v]) : F_INF;
        if (p1 < thr[v]) {
          unsigned pos = atomicAdd(&s_cnt[m], 1u);
          s_pd[m * CAP + pos] = p1;
          s_pi[m * CAP + pos] = (unsigned)c1;
        }
      }
    }
    __syncthreads();

    // streaming flush: wave wv owns rows 2wv, 2wv+1
    for (int rr = 0; rr < 2; ++rr) {
      int row = wv * 2 + rr;
      int cnt = (int)s_cnt[row];
      if (cnt > 128) {                     // keeps pool <= CAP next iter
        float pd[12]; unsigned pi[12];
#pragma unroll
        for (int j = 0; j < 12; ++j) {
          int slot = lane + 32 * j;
          bool ok = slot < cnt;
          pd[j] = ok ? s_pd[row * CAP + slot] : F_INF;
          pi[j] = ok ? s_pi[row * CAP + slot] : 0u;
        }
        float thr = F_INF;
        for (int i = 0; i < K_NB; ++i) {   // 50x argmin-extract
          float lv = F_INF; int lj = 0;
#pragma unroll
          for (int j = 0; j < 12; ++j) if (pd[j] < lv) { lv = pd[j]; lj = j; }
          float v = lv; int who = lane;
          for (int off = 16; off > 0; off >>= 1) {
            float ov = __shfl_xor(v, off, 32);
            int   ow = __shfl_xor(who, off, 32);
            if (ov < v || (ov == v && ow < who)) { v = ov; who = ow; }
          }
          if (lane == who) {
            s_pd[row * CAP + i] = pd[lj];
            s_pi[row * CAP + i] = pi[lj];
            pd[lj] = F_INF;
          }
          thr = v;
        }
        if (lane == 0) { s_cnt[row] = K_NB; s_thr[row] = thr; }
      }
    }
    __syncthreads();
  }

  // final: exact top-50 per row, write to workspace (pad with +inf)
  for (int rr = 0; rr < 2; ++rr) {
    int row = wv * 2 + rr;
    int cnt = (int)s_cnt[row];
    float pd[12]; unsigned pi[12];
#pragma unroll
    for (int j = 0; j < 12; ++j) {
      int slot = lane + 32 * j;
      bool ok = slot < cnt;
      pd[j] = ok ? s_pd[row * CAP + slot] : F_INF;
      pi[j] = ok ? s_pi[row * CAP + slot] : 0u;
    }
    size_t outBase = ((size_t)(r0 + row) * NSPLIT + split) * K_NB;
    for (int i = 0; i < K_NB; ++i) {
      float lv = F_INF; int lj = 0;
#pragma unroll
      for (int j = 0; j < 12; ++j) if (pd[j] < lv) { lv = pd[j]; lj = j; }
      float v = lv; int who = lane;
      for (int off = 16; off > 0; off >>= 1) {
        float ov = __shfl_xor(v, off, 32);
        int   ow = __shfl_xor(who, off, 32);
        if (ov < v || (ov == v && ow < who)) { v = ov; who = ow; }
      }
      if (lane == who) {
        cand_d[outBase + i] = pd[lj];
        cand_i[outBase + i] = pi[lj];
        pd[lj] = F_INF;
      }
    }
  }
}

// ---------------------------------------------------------------------------
// Kernel 4: per-row merge of 8*50 candidates -> exact top-50 -> value
// one wave per row; 128 blocks of 256 threads
// ---------------------------------------------------------------------------
__global__ void __launch_bounds__(256) k_merge(
    const float* __restrict__ h, const float* __restrict__ vals,
    const float* __restrict__ cand_d, const unsigned* __restrict__ cand_i,
    float* __restrict__ value) {
  int lane = threadIdx.x & 31;
  int row  = blockIdx.x * 8 + (threadIdx.x >> 5);

  // h_sq for this row
  float a0 = h[(size_t)row * H_DIM + lane];
  float a1 = h[(size_t)row * H_DIM + 32 + lane];
  float hsq = a0 * a0 + a1 * a1;
  for (int off = 16; off > 0; off >>= 1) hsq += __shfl_xor(hsq, off, 32);

  const int TOT = NSPLIT * K_NB;   // 400
  float pd[13]; unsigned pi[13];
  size_t base = (size_t)row * TOT;
#pragma unroll
  for (int j = 0; j < 13; ++j) {
    int slot = lane + 32 * j;
    bool ok = slot < TOT;
    pd[j] = ok ? cand_d[base + slot] : F_INF;
    pi[j] = ok ? cand_i[base + slot] : 0u;
  }

  float accW = 0.f, accWV = 0.f;
  for (int i = 0; i < K_NB; ++i) {
    float lv = F_INF; int lj = 0;
#pragma unroll
    for (int j = 0; j < 13; ++j) if (pd[j] < lv) { lv = pd[j]; lj = j; }
    float v = lv; int who = lane;
    for (int off = 16; off > 0; off >>= 1) {
      float ov = __shfl_xor(v, off, 32);
      int   ow = __shfl_xor(who, off, 32);
      if (ov < v || (ov == v && ow < who)) { v = ov; who = ow; }
    }
    if (lane == who) {
      float d = pd[lj] + hsq;           // add deferred per-row constant
      d = d > 0.f ? d : 0.f;            // clamp fp noise (matches reference)
      float w = 1.f / (d + DND_DELTA);  // inf-pad -> w == 0, harmless
      accW  += w;
      accWV += w * vals[pi[lj]];
      pd[lj] = F_INF;
    }
  }
  for (int off = 16; off > 0; off >>= 1) {
    accW  += __shfl_xor(accW, off, 32);
    accWV += __shfl_xor(accWV, off, 32);
  }
  if (lane == 0) value[row] = accWV / accW;
}

// ---------------------------------------------------------------------------
extern "C" void kernel_launch(void* const* d_in, const int* in_sizes, int n_in,
                              void* d_out, int out_size, void* d_ws, size_t ws_size,
                              hipStream_t stream) {
  const float* x    = (const float*)d_in[0];
  const float* W1   = (const float*)d_in[1];
  const float* b1   = (const float*)d_in[2];
  const float* Wp   = (const float*)d_in[3];
  const float* bp   = (const float*)d_in[4];
  const float* keys = (const float*)d_in[5];
  const float* vals = (const float*)d_in[6];

  float* pol   = (float*)d_out;                            // [1024,18]
  float* value = (float*)d_out + B_ROWS * A_DIM;           // [1024,1]
  float* h     = (float*)d_out + B_ROWS * A_DIM + B_ROWS;  // [1024,64]

  const size_t nCand = (size_t)B_ROWS * NSPLIT * K_NB;     // 409600
  float*    cand_d = (float*)d_ws;
  unsigned* cand_i = (unsigned*)((char*)d_ws + nCand * sizeof(float));

  k_hidden<<<(B_ROWS / 16) * 4 / 8, 256, 0, stream>>>(x, W1, b1, h);
  k_policy<<<(B_ROWS * A_DIM + 255) / 256, 256, 0, stream>>>(h, Wp, bp, pol);
  k_dnd_partial<<<(B_ROWS / 16) * NSPLIT, 256, 0, stream>>>(h, keys, cand_d, cand_i);
  k_merge<<<B_ROWS / 8, 256, 0, stream>>>(h, vals, cand_d, cand_i, value);
}